// Hadamard_40664750358795
// MI455X (gfx1250) — compile-verified
//
#include <hip/hip_runtime.h>

// ---------------------------------------------------------------------------
// Batched 4096-point Walsh-Hadamard transform, fp32, scale 1/64 = 1/sqrt(4096).
//
// H4096 = H16 (x) H16 (x) H16  -> three passes of a 16x16 (+/-1) GEMM done with
// chained V_WMMA_F32_16X16X4_F32 (fp32 WMMA, K=4): exact fp32 arithmetic.
//
// Memory bound: 512 MB @ 23.3 TB/s ~= 22 us floor. One wave owns one row; the
// row lives in LDS (padded layout, bank-conflict-free, immediate ds offsets)
// across all three stages. Global<->LDS traffic uses CDNA5 async-to-LDS ops
// (ASYNCcnt) when the toolchain exposes them.
// ---------------------------------------------------------------------------

typedef float v2f __attribute__((ext_vector_type(2)));
typedef float v8f __attribute__((ext_vector_type(8)));
typedef int   v4i __attribute__((ext_vector_type(4)));

// pointer types for the async-to-LDS builtins: (global v4i*, lds v4i*, imm, imm)
typedef v4i __attribute__((address_space(1))) *gv4i_p;   // global
typedef v4i __attribute__((address_space(3))) *lv4i_p;   // LDS

#define FWHT_DIM 4096
#define WAVES_PER_BLOCK 2

// Padded LDS layout: addr(a,b,c) = a*PITCH_A + b*PITCH_B + c   (floats)
//  - PITCH_B = 20 (= 16+4):  stride-20 lane patterns hit 16 distinct banks
//  - PITCH_A = 16*20+4 = 324: stride-324 (stage-3 stores) ≡ 4 mod 64 -> 16 banks
//  - both pitches ≡ 0 mod 4 -> every c-aligned float4 stays 16B-aligned
#define PITCH_B 20
#define PITCH_A 324
#define WAVE_FLOATS 5184   // > 15*324+15*20+16, rounded to multiple of 16

#if defined(__gfx1250__) && __has_builtin(__builtin_amdgcn_global_load_async_to_lds_b128) && \
    __has_builtin(__builtin_amdgcn_global_store_async_from_lds_b128)
#define USE_ASYNC_LDS 1
#else
#define USE_ASYNC_LDS 0
#endif

// Sylvester Hadamard entry with magnitude folded in:
//   value = +/- mag,  sign = popcount(i&j) parity
__device__ __forceinline__ float hs(int i, int j, unsigned mag) {
    return __int_as_float(mag | ((unsigned)(__popc(i & j) & 1) << 31));
}

// dense row index -> padded LDS index (floats)
__device__ __forceinline__ int pad_idx(int idx) {
    return idx + (idx >> 8) * 68 + ((idx >> 4) & 15) * 4;
}

__global__ __launch_bounds__(WAVES_PER_BLOCK * 32)
void fwht4096_wmma_kernel(const float* __restrict__ x, float* __restrict__ y) {
    __shared__ float lds[WAVES_PER_BLOCK][WAVE_FLOATS];   // ~40.5 KB / block

    const int lane = threadIdx.x & 31;
    const int wave = threadIdx.x >> 5;
    float* buf = lds[wave];

    const long long row = (long long)blockIdx.x * WAVES_PER_BLOCK + wave;
    const float* __restrict__ src = x + row * FWHT_DIM;
    float*       __restrict__ dst = y + row * FWHT_DIM;

    // ---------------- global -> LDS (async b128, coalesced) ----------------
    #pragma unroll
    for (int i = 0; i < FWHT_DIM / (32 * 4); ++i) {        // 32 iters
        const int idx = (i * 32 + lane) * 4;
        const int p   = pad_idx(idx);
#if USE_ASYNC_LDS
        __builtin_amdgcn_global_load_async_to_lds_b128(
            (gv4i_p)(src + idx), (lv4i_p)(buf + p), 0, 0);
#else
        const float4 v = *(const float4*)(src + idx);
        *(float4*)(buf + p) = v;
#endif
    }
#if USE_ASYNC_LDS
    asm volatile("s_wait_asynccnt 0x0" ::: "memory");
#endif
    __syncthreads();

    const int half = lane >> 4;    // which half-wave
    const int nm   = lane & 15;    // A-row (M) / B,C,D-column (N)

    // B operands: H16 sliced into 4 chunks of K=4 for f32 WMMA.
    // f32 4x16 B layout: lane l, vgpr j holds B[2*(l>>4)+j][l&15].
    v2f B1[4];   // entries +/-1        (stages 1,2)
    v2f B3[4];   // entries +/-(1/64)   (stage 3, scale folded in)
    #pragma unroll
    for (int k0 = 0; k0 < 4; ++k0) {
        const int kb = k0 * 4 + half * 2;
        B1[k0].x = hs(kb,     nm, 0x3F800000u);   // +/-1.0
        B1[k0].y = hs(kb + 1, nm, 0x3F800000u);
        B3[k0].x = hs(kb,     nm, 0x3C800000u);   // +/-(1/64)
        B3[k0].y = hs(kb + 1, nm, 0x3C800000u);
    }

    // ============ stage 1: contract c'  (tile t = a; A rows = b) ============
    // D[b][c] = sum_c' x[t,b,c'] * H[c'][c]    (in place in segment t)
    #pragma unroll
    for (int t = 0; t < 16; ++t) {
        v8f acc = {};
        #pragma unroll
        for (int k0 = 0; k0 < 4; ++k0) {
            const int ka = k0 * 4 + half * 2;               // even -> b64
            const v2f a = *(const v2f*)(buf + t * PITCH_A + nm * PITCH_B + ka);
            acc = __builtin_amdgcn_wmma_f32_16x16x4_f32(
                      false, a, false, B1[k0], (short)0, acc, false, false);
        }
        #pragma unroll
        for (int r = 0; r < 8; ++r)
            buf[t * PITCH_A + (r + 8 * half) * PITCH_B + nm] = acc[r];
    }
    __syncthreads();

    // ============ stage 2: contract b'  (tile t = a; A rows = c) ============
    // D[c][b] = sum_b' T1[t,b',c] * H[b'][b]   (in place in segment t)
    #pragma unroll
    for (int t = 0; t < 16; ++t) {
        v8f acc = {};
        #pragma unroll
        for (int k0 = 0; k0 < 4; ++k0) {
            const int ka = k0 * 4 + half * 2;
            v2f a;
            a.x = buf[t * PITCH_A + (ka    ) * PITCH_B + nm];
            a.y = buf[t * PITCH_A + (ka + 1) * PITCH_B + nm];
            acc = __builtin_amdgcn_wmma_f32_16x16x4_f32(
                      false, a, false, B1[k0], (short)0, acc, false, false);
        }
        // rows r are contiguous here -> two 16B-aligned b128 stores
        float4 lo = make_float4(acc[0], acc[1], acc[2], acc[3]);
        float4 hi = make_float4(acc[4], acc[5], acc[6], acc[7]);
        float* st = buf + t * PITCH_A + nm * PITCH_B + 8 * half;
        *(float4*)(st)     = lo;
        *(float4*)(st + 4) = hi;
    }
    __syncthreads();

    // ============ stage 3: contract a'  (tile t = c; A rows = b) ============
    // D[b][a] = sum_a' T2[a',b,t] * (H[a'][a]/64)
    // Tile t only touches addresses with c == t -> in place is safe.
    #pragma unroll
    for (int t = 0; t < 16; ++t) {
        v8f acc = {};
        #pragma unroll
        for (int k0 = 0; k0 < 4; ++k0) {
            const int ka = k0 * 4 + half * 2;
            v2f a;
            a.x = buf[(ka    ) * PITCH_A + nm * PITCH_B + t];
            a.y = buf[(ka + 1) * PITCH_A + nm * PITCH_B + t];
            acc = __builtin_amdgcn_wmma_f32_16x16x4_f32(
                      false, a, false, B3[k0], (short)0, acc, false, false);
        }
        #pragma unroll
        for (int r = 0; r < 8; ++r)
            buf[nm * PITCH_A + (r + 8 * half) * PITCH_B + t] = acc[r];
    }
    __syncthreads();

    // ---------------- LDS -> global (async b128, coalesced) ----------------
    #pragma unroll
    for (int i = 0; i < FWHT_DIM / (32 * 4); ++i) {
        const int idx = (i * 32 + lane) * 4;
        const int p   = pad_idx(idx);
#if USE_ASYNC_LDS
        __builtin_amdgcn_global_store_async_from_lds_b128(
            (gv4i_p)(dst + idx), (lv4i_p)(buf + p), 0, 0);
#else
        const float4 v = *(const float4*)(buf + p);
        *(float4*)(dst + idx) = v;
#endif
    }
#if USE_ASYNC_LDS
    asm volatile("s_wait_asynccnt 0x0" ::: "memory");
#endif
}

extern "C" void kernel_launch(void* const* d_in, const int* in_sizes, int n_in,
                              void* d_out, int out_size, void* d_ws, size_t ws_size,
                              hipStream_t stream) {
    (void)n_in; (void)d_ws; (void)ws_size; (void)out_size;
    const float* x = (const float*)d_in[0];
    float*       y = (float*)d_out;

    const int rows   = in_sizes[0] / FWHT_DIM;            // 16384
    const int blocks = rows / WAVES_PER_BLOCK;            // 8192 (exact)
    fwht4096_wmma_kernel<<<blocks, WAVES_PER_BLOCK * 32, 0, stream>>>(x, y);
}